// SubgraphingLayer_90692529422534
// MI455X (gfx1250) — compile-verified
//
#include <hip/hip_runtime.h>
#include <hip/hip_bf16.h>
#include <stdint.h>

#define BATCH 32
#define NNODE 256
#define NEDGE 8
#define NFEAT 16
#define K1V   32

typedef float v4f __attribute__((ext_vector_type(4)));

// ---------------------------------------------------------------------------
// Kernel 1: one wave32 per (b, s) ordered BFS collecting K1 vertices.
// Matches reference: FIFO queue, neighbors enqueued in ascending node index,
// dedup vs seen; queue slots default to 0 (node 0) if exhausted (head < 32).
// ---------------------------------------------------------------------------
__global__ __launch_bounds__(256) void bfs_kernel(const float* __restrict__ adj,
                                                  int* __restrict__ proc_out) {
    __shared__ int queue[8][NNODE];   // 8 waves per block, 1KB each

    const int lane = threadIdx.x & 31;
    const int wv   = threadIdx.x >> 5;
    const int idx  = blockIdx.x * 8 + wv;          // 0 .. B*N-1
    const int b    = idx >> 8;                     // / NNODE
    const int s    = idx & (NNODE - 1);

    int* q = &queue[wv][0];
    // head only ever reaches 31, so zero-init the first 32 slots (ref queue=0)
    q[lane] = 0;
    if (lane == 0) q[0] = s;

    // lane owns seen bits for nodes [lane*8, lane*8+8)
    unsigned seen = 0;
    if ((s >> 3) == lane) seen = 1u << (s & 7);

    const float* adjB = adj + (size_t)b * NNODE * NNODE;

    int head = 0, tail = 1;
    int myproc = 0;

    #pragma unroll 1
    for (int it = 0; it < K1V; ++it) {
        const int cur = q[head];                   // uniform across wave
        if (lane == it) myproc = cur;

        const v4f* row = (const v4f*)(adjB + (size_t)cur * NNODE + lane * 8);
        const v4f a0 = row[0];
        const v4f a1 = row[1];

        unsigned m = 0;
        m |= (unsigned)(a0.x != 0.0f) << 0;
        m |= (unsigned)(a0.y != 0.0f) << 1;
        m |= (unsigned)(a0.z != 0.0f) << 2;
        m |= (unsigned)(a0.w != 0.0f) << 3;
        m |= (unsigned)(a1.x != 0.0f) << 4;
        m |= (unsigned)(a1.y != 0.0f) << 5;
        m |= (unsigned)(a1.z != 0.0f) << 6;
        m |= (unsigned)(a1.w != 0.0f) << 7;
        m &= ~seen;
        seen |= m;

        const int cnt = __popc(m);
        // inclusive wave prefix sum (wave32)
        int pref = cnt;
        #pragma unroll
        for (int off = 1; off < 32; off <<= 1) {
            int t = __shfl_up(pref, off, 32);
            if (lane >= off) pref += t;
        }
        const int total = __shfl(pref, 31, 32);
        int pos = tail + (pref - cnt);             // exclusive base for my bits

        unsigned mm = m;
        while (mm) {
            const int bit = __builtin_ctz(mm);
            q[pos++] = lane * 8 + bit;             // ascending node order
            mm &= mm - 1;
        }
        tail += total;
        ++head;
    }

    proc_out[(size_t)idx * K1V + lane] = myproc;   // coalesced; re-read soon (RT)
}

// ---------------------------------------------------------------------------
// Kernel 2: one 256-thread block per (b, s). Stage the 32 selected adjacency
// rows into LDS with CDNA5 async global->LDS copies (overlapped with the
// feature gather), then emit sub_adj / sub_edges / sub_feats with NT stores
// so the 319 MB output stream does not evict the L2-resident inputs.
// ---------------------------------------------------------------------------
__global__ __launch_bounds__(256) void gather_kernel(
    const float* __restrict__ adj, const float* __restrict__ edges,
    const float* __restrict__ feats, const int* __restrict__ proc,
    float* __restrict__ out_adj, float* __restrict__ out_edges,
    float* __restrict__ out_feats) {
    __shared__ int   p[K1V];
    __shared__ __align__(16) float adjTile[K1V * NNODE];   // 32 KB

    const int tid  = threadIdx.x;
    const int lane = tid & 31;
    const int wv   = tid >> 5;
    const int bs   = blockIdx.x;       // 0 .. B*N-1
    const int b    = bs >> 8;

    if (tid < K1V) p[tid] = proc[(size_t)bs * K1V + tid];
    __syncthreads();

    const float* adjB = adj + (size_t)b * NNODE * NNODE;

    // Async-stage 32 adjacency rows (1KB each). 8 waves x 4 rows x 2 B128 ops.
    #pragma unroll
    for (int rr = 0; rr < 4; ++rr) {
        const int row  = wv * 4 + rr;
        const int node = p[row];
        const float* g  = adjB + (size_t)node * NNODE + lane * 4;   // 16B/lane
        const float* g2 = g + 128;
        uint32_t l  = (uint32_t)(uintptr_t)&adjTile[row * NNODE + lane * 4];
        uint32_t l2 = l + 512;
        asm volatile("global_load_async_to_lds_b128 %0, %1, off"
                     :: "v"(l), "v"(g) : "memory");
        asm volatile("global_load_async_to_lds_b128 %0, %1, off"
                     :: "v"(l2), "v"(g2) : "memory");
    }

    // Overlap: feature gather (32 x 16 floats) while async copies fly.
    {
        float* outF = out_feats + (size_t)bs * (K1V * NFEAT);
        const float* fB = feats + (size_t)b * NNODE * NFEAT;
        #pragma unroll
        for (int k = tid; k < K1V * NFEAT; k += 256) {
            const int i = k >> 4;
            const int f = k & 15;
            __builtin_nontemporal_store(fB[(size_t)p[i] * NFEAT + f], &outF[k]);
        }
    }

    asm volatile("s_wait_asynccnt 0x0" ::: "memory");
    __syncthreads();

    // sub_adj + masked sub_edges: 1024 (i,j) pairs, 4 per thread.
    const float* edgB = edges + (size_t)b * NNODE * NNODE * NEDGE;
    float* oA = out_adj   + (size_t)bs * (K1V * K1V);
    float* oE = out_edges + (size_t)bs * (K1V * K1V * NEDGE);

    #pragma unroll
    for (int pr = tid; pr < K1V * K1V; pr += 256) {
        const int i = pr >> 5;
        const int j = pr & 31;
        const float a = adjTile[i * NNODE + p[j]];
        __builtin_nontemporal_store(a, &oA[pr]);

        v4f e0 = (v4f)(0.0f);
        v4f e1 = (v4f)(0.0f);
        if (a != 0.0f) {
            const v4f* ev = (const v4f*)(edgB +
                ((size_t)p[i] * NNODE + (size_t)p[j]) * NEDGE);
            e0 = ev[0];
            e1 = ev[1];
        }
        v4f* dst = (v4f*)(oE + (size_t)pr * NEDGE);
        __builtin_nontemporal_store(e0, &dst[0]);
        __builtin_nontemporal_store(e1, &dst[1]);
    }
}

// ---------------------------------------------------------------------------
extern "C" void kernel_launch(void* const* d_in, const int* in_sizes, int n_in,
                              void* d_out, int out_size, void* d_ws, size_t ws_size,
                              hipStream_t stream) {
    const float* adj   = (const float*)d_in[0];   // (B, N, N)
    const float* edges = (const float*)d_in[1];   // (B, N, N, E)
    const float* feats = (const float*)d_in[2];   // (B, N, F)

    float* out_adj   = (float*)d_out;                                   // B*N*32*32
    float* out_edges = out_adj + (size_t)BATCH * NNODE * K1V * K1V;     // B*N*32*32*8
    float* out_feats = out_edges + (size_t)BATCH * NNODE * K1V * K1V * NEDGE;

    int* proc = (int*)d_ws;   // B*N*K1 int32 = 1 MB

    // Phase 1: BFS — 8 waves (one per (b,s)) per 256-thread block.
    bfs_kernel<<<(BATCH * NNODE) / 8, 256, 0, stream>>>(adj, proc);

    // Phase 2: gather — one block per (b,s).
    gather_kernel<<<BATCH * NNODE, 256, 0, stream>>>(
        adj, edges, feats, proc, out_adj, out_edges, out_feats);
}